// A_Snn_85237920956577
// MI455X (gfx1250) — compile-verified
//
#include <hip/hip_runtime.h>
#include <hip/hip_bf16.h>
#include <math.h>

typedef __attribute__((ext_vector_type(2))) float v2f;
typedef __attribute__((ext_vector_type(8))) float v8f;

#define B_ 64
#define T_ 512
#define I_ 1024
#define O_ 1024
#define A_ 3
#define NWG 64   // scan grid: 64 workgroups, each owns 16 output columns

__device__ __forceinline__ float sigmoidf_(float x) { return 1.f / (1.f + __expf(-x)); }
__device__ __forceinline__ float softplusf_(float x) {
  return fmaxf(x, 0.f) + log1pf(__expf(-fabsf(x)));
}

// ---------------------------------------------------------------------------
// Init: zero h ping buffer + grid-barrier counter (d_ws is poisoned 0xAA).
// ---------------------------------------------------------------------------
__global__ void __launch_bounds__(256) init_kernel(float* h0, unsigned* bar) {
  int i = blockIdx.x * 256 + threadIdx.x;
  if (i < B_ * O_) h0[i] = 0.f;
  if (i == 0) *bar = 0u;
}

// ---------------------------------------------------------------------------
// Projection GEMM: OUT[t*64+b][o] = sum_i x[b][t][i] * W[o][i] + bias[o]
// grid = (O/64, T, 2)  z selects (ww_x,bw,xw) vs (wb_x,bb,xb)
// 64x64 output tile per WG, fp32 WMMA 16x16x4, K staged through LDS.
// ---------------------------------------------------------------------------
__global__ void __launch_bounds__(256) proj_kernel(
    const float* __restrict__ x,
    const float* __restrict__ wwx, const float* __restrict__ bw,
    const float* __restrict__ wbx, const float* __restrict__ bb,
    float* __restrict__ xw, float* __restrict__ xb) {
  __shared__ float Xs[64][68];
  __shared__ float Ws[64][68];
  const int t = blockIdx.y;
  const int o0 = blockIdx.x * 64;
  const float* __restrict__ W    = blockIdx.z ? wbx : wwx;
  const float* __restrict__ bias = blockIdx.z ? bb : bw;
  float* __restrict__ out        = blockIdx.z ? xb : xw;

  const int tid = threadIdx.x;
  const int lane = tid & 31;
  const int wid  = tid >> 5;
  const int ln = lane & 15, hi = lane >> 4;
  const int mi = wid & 3;            // 16-row subtile
  const int nb = (wid >> 2) * 2;     // pair of 16-col subtiles

  v8f acc0 = {}; v8f acc1 = {};

  for (int k0 = 0; k0 < I_; k0 += 64) {
    // stage X tile (rows b=0..63) and W tile (rows o=o0..o0+63), 64 K each
    const int lrow = tid >> 4;
    const int lkk  = (tid & 15) * 4;
#pragma unroll
    for (int rep = 0; rep < 4; ++rep) {
      int row = lrow + rep * 16;
      *(float4*)&Xs[row][lkk] =
          *(const float4*)(x + ((size_t)row * T_ + t) * I_ + k0 + lkk);
      *(float4*)&Ws[row][lkk] =
          *(const float4*)(W + (size_t)(o0 + row) * I_ + k0 + lkk);
    }
    __syncthreads();
#pragma unroll
    for (int ks = 0; ks < 64; ks += 4) {
      v2f a  = *(const v2f*)&Xs[mi * 16 + ln][ks + 2 * hi];
      v2f b0 = *(const v2f*)&Ws[nb * 16 + ln][ks + 2 * hi];
      v2f b1 = *(const v2f*)&Ws[(nb + 1) * 16 + ln][ks + 2 * hi];
      acc0 = __builtin_amdgcn_wmma_f32_16x16x4_f32(false, a, false, b0,
                                                   (short)0, acc0, false, false);
      acc1 = __builtin_amdgcn_wmma_f32_16x16x4_f32(false, a, false, b1,
                                                   (short)0, acc1, false, false);
    }
    __syncthreads();
  }
#pragma unroll
  for (int r = 0; r < 8; ++r) {
    int b = mi * 16 + r + 8 * hi;                 // row within 64-row tile
    size_t robase = ((size_t)t * B_ + b) * O_;
    int oc0 = o0 + nb * 16 + ln;
    int oc1 = o0 + (nb + 1) * 16 + ln;
    out[robase + oc0] = acc0[r] + bias[oc0];
    out[robase + oc1] = acc1[r] + bias[oc1];
  }
}

// ---------------------------------------------------------------------------
// Tiny A=3 projection: xa[t][b][a] = ba[a] + dot(x[b][t][:], wa_x[a][:])
// one wave per (t,b,a) dot product.
// ---------------------------------------------------------------------------
__global__ void __launch_bounds__(256) proja_kernel(
    const float* __restrict__ x, const float* __restrict__ wax,
    const float* __restrict__ ba, float* __restrict__ xa) {
  const int lane = threadIdx.x & 31;
  const int gw = blockIdx.x * 8 + (threadIdx.x >> 5);
  if (gw >= T_ * B_ * A_) return;
  const int a = gw % A_;
  const int rb = gw / A_;
  const int b = rb % B_;
  const int t = rb / B_;
  const float* xr = x + ((size_t)b * T_ + t) * I_;
  const float* wr = wax + (size_t)a * I_;
  float s = 0.f;
#pragma unroll 4
  for (int k = lane; k < I_; k += 32) s += xr[k] * wr[k];
#pragma unroll
  for (int off = 16; off; off >>= 1) s += __shfl_down(s, off, 32);
  if (lane == 0) xa[((size_t)t * B_ + b) * A_ + a] = s + ba[a];
}

// ---------------------------------------------------------------------------
// Persistent scan kernel: 64 WGs x 256 threads, one grid barrier per step.
// WG owns 16 output columns o0..o0+15. Waves 0-3: W-gate rows, waves 4-7:
// B-gate rows.  y slice lives in registers of waves 0-3 for all 512 steps.
// ---------------------------------------------------------------------------
__global__ void __launch_bounds__(256, 1) scan_kernel(
    const float* __restrict__ xw, const float* __restrict__ xb,
    const float* __restrict__ xa,
    const float* __restrict__ wwy, const float* __restrict__ wby,
    const float* __restrict__ way,
    float* __restrict__ h0, float* __restrict__ h1,
    float* __restrict__ out, float* __restrict__ yfin,
    unsigned* __restrict__ bar) {
  __shared__ float Hs[64][68];        // h chunk  [b=64][k=64]
  __shared__ float Wts[2][16][68];    // ww_y / wb_y slices [o=16][k=64]
  __shared__ float Zb[64][16];        // tanh-gate pre-activations exchange
  __shared__ float Wa[64][4];         // per-batch attention weights
  __shared__ float WayS[A_][O_];      // wa_y staged once (12 KB)

  const int tid = threadIdx.x;
  const int lane = tid & 31;
  const int wid  = tid >> 5;
  const int ln = lane & 15, hi = lane >> 4;
  const int o0 = blockIdx.x * 16;
  const int mat = wid >> 2;           // 0 = sigmoid gate (ww_y), 1 = tanh gate (wb_y)
  const int m0 = (wid & 3) * 16;      // 16-row subtile of b

  for (int i = tid; i < A_ * O_; i += 256) WayS[i >> 10][i & (O_ - 1)] = way[i];

  const int lrow = tid >> 4;
  const int lkk  = (tid & 15) * 4;

  v8f yv = {};                        // y[b][o] slice (waves 0-3)

  for (int t = 0; t < T_; ++t) {
    const float* hprev = (t & 1) ? h1 : h0;
    float* hnext       = (t & 1) ? h0 : h1;
    v8f acc = {};
    // ---- z = h_prev @ Wy^T over K=1024 in 16 chunks of 64 ----
    for (int c = 0; c < 16; ++c) {
      const int k0 = c * 64;
#pragma unroll
      for (int rep = 0; rep < 4; ++rep) {
        int row = lrow + rep * 16;
        *(float4*)&Hs[row][lkk] =
            *(const float4*)(hprev + (size_t)row * O_ + k0 + lkk);
      }
      *(float4*)&Wts[0][lrow][lkk] =
          *(const float4*)(wwy + (size_t)(o0 + lrow) * O_ + k0 + lkk);
      *(float4*)&Wts[1][lrow][lkk] =
          *(const float4*)(wby + (size_t)(o0 + lrow) * O_ + k0 + lkk);
      __syncthreads();
#pragma unroll
      for (int ks = 0; ks < 64; ks += 4) {
        v2f a = *(const v2f*)&Hs[m0 + ln][ks + 2 * hi];
        v2f bf = *(const v2f*)&Wts[mat][ln][ks + 2 * hi];
        acc = __builtin_amdgcn_wmma_f32_16x16x4_f32(false, a, false, bf,
                                                    (short)0, acc, false, false);
      }
      __syncthreads();
    }
    // ---- wa[b][a] = sigmoid(xa + h_prev . wa_y[a]) : 192 dots, 24/wave ----
    const float* xarow = xa + (size_t)t * B_ * A_;
#pragma unroll 2
    for (int j = 0; j < 24; ++j) {
      int p = wid * 24 + j;
      int b = p / A_, a = p % A_;
      const float* hr = hprev + (size_t)b * O_;
      float s = 0.f;
#pragma unroll 4
      for (int k = lane; k < O_; k += 32) s += hr[k] * WayS[a][k];
#pragma unroll
      for (int off = 16; off; off >>= 1) s += __shfl_down(s, off, 32);
      if (lane == 0) Wa[b][a] = sigmoidf_(xarow[b * A_ + a] + s);
    }
    // ---- add precomputed input projection; exchange tanh-gate via LDS ----
    const size_t xoff = (size_t)t * B_ * O_;
    if (mat == 1) {
#pragma unroll
      for (int r = 0; r < 8; ++r) {
        int b = m0 + r + 8 * hi;
        Zb[b][ln] = acc[r] + xb[xoff + (size_t)b * O_ + o0 + ln];
      }
    }
    __syncthreads();
    if (mat == 0) {
#pragma unroll
      for (int r = 0; r < 8; ++r) {
        int b = m0 + r + 8 * hi;
        float zw = acc[r] + xw[xoff + (size_t)b * O_ + o0 + ln];
        float w  = sigmoidf_(zw);
        float bv = tanhf(Zb[b][ln]);
        float y  = w * yv[r] + bv;
        y = (y >= 0.f) ? y : 0.01f * y;      // leaky relu (in-place in ref)
        yv[r] = y;
        float th = tanhf(y);
        float sp = softplusf_(y);
        float h  = tanhf((y * Wa[b][0] + th * Wa[b][1] + sp * Wa[b][2]) * 0.1f);
        hnext[(size_t)b * O_ + o0 + ln] = h;
        out[((size_t)b * T_ + t) * O_ + o0 + ln] = h;   // hs -> [B,T,O]
      }
    }
    // prefetch next step's projections into cache (global_prefetch_b8)
    if (t + 1 < T_) {
      const size_t nxt = (size_t)(t + 1) * B_ * O_ + (size_t)(tid >> 2) * O_ +
                         o0 + (tid & 3) * 4;
      __builtin_prefetch(xw + nxt, 0, 1);
      __builtin_prefetch(xb + nxt, 0, 1);
    }
    // ---- device-wide barrier: release h slice, acquire everyone else's ----
    __syncthreads();
    if (tid == 0) {
      __hip_atomic_fetch_add(bar, 1u, __ATOMIC_RELEASE, __HIP_MEMORY_SCOPE_AGENT);
      const unsigned target = (unsigned)(t + 1) * (unsigned)NWG;
      while (__hip_atomic_load(bar, __ATOMIC_ACQUIRE, __HIP_MEMORY_SCOPE_AGENT) <
             target)
        __builtin_amdgcn_s_sleep(2);
    }
    __syncthreads();
    __threadfence();   // agent-scope acquire in every wave (WGP$ inv)
  }
  // ---- y_fin ----
  if (mat == 0) {
#pragma unroll
    for (int r = 0; r < 8; ++r) {
      int b = m0 + r + 8 * hi;
      yfin[(size_t)b * O_ + o0 + ln] = yv[r];
    }
  }
}

// ---------------------------------------------------------------------------
extern "C" void kernel_launch(void* const* d_in, const int* in_sizes, int n_in,
                              void* d_out, int out_size, void* d_ws, size_t ws_size,
                              hipStream_t stream) {
  (void)in_sizes; (void)n_in; (void)out_size; (void)ws_size;
  const float* x   = (const float*)d_in[0];
  const float* wwx = (const float*)d_in[1];
  const float* wwy = (const float*)d_in[2];
  const float* bw  = (const float*)d_in[3];
  const float* wbx = (const float*)d_in[4];
  const float* wby = (const float*)d_in[5];
  const float* bb  = (const float*)d_in[6];
  const float* wax = (const float*)d_in[7];
  const float* way = (const float*)d_in[8];
  const float* ba  = (const float*)d_in[9];

  float* out  = (float*)d_out;                       // [B,T,O]
  float* yfin = out + (size_t)B_ * T_ * O_;          // [B,O]

  const size_t NP = (size_t)T_ * B_ * O_;            // 33,554,432 floats
  float* ws = (float*)d_ws;
  float* xw = ws;
  float* xb = xw + NP;
  float* xa = xb + NP;                               // T*B*A
  float* h0 = xa + (size_t)T_ * B_ * A_;
  float* h1 = h0 + (size_t)B_ * O_;
  unsigned* bar = (unsigned*)(h1 + (size_t)B_ * O_);

  init_kernel<<<256, 256, 0, stream>>>(h0, bar);
  proj_kernel<<<dim3(O_ / 64, T_, 2), 256, 0, stream>>>(x, wwx, bw, wbx, bb, xw, xb);
  proja_kernel<<<(T_ * B_ * A_) / 8, 256, 0, stream>>>(x, wax, ba, xa);
  scan_kernel<<<NWG, 256, 0, stream>>>(xw, xb, xa, wwy, wby, way, h0, h1, out,
                                       yfin, bar);
}